// InfoGCN_73985106641331
// MI455X (gfx1250) — compile-verified
//
#include <hip/hip_runtime.h>

// ---------------------------------------------------------------------------
// InfoGCN block, fused for MI455X (gfx1250, wave32, WMMA + TDM).
//
// Exact algebraic collapse of the reference:
//   X64[n,ci,b,v]  = sum_{i<64} x[n,ci,64b+i,v]                  (252 vals/n)
//   s[j,k,v]       = sum_ci conv_w[kc,ci]*X64[ci,u&3,v] + 64*conv_b[kc],
//                    u = 3j+k, kc = u>>2
//   y2[row,k]      = sum_v s[j,k,v]*A[k,v,w],  row = j*21+w
//   y3[row,co]     = b_emb[co] + pos_emb[w,co] + sum_k y2[row,k]*W_emb[co,k]
//                    -> V_WMMA_F32_16X16X4_F32 (K=3 padded to 4)
//   BN channel     = row>>2 (per-n flat p = row*64+co, ch = p>>8)
// Output memory layout == y3 memory layout (all reshapes are raw views).
// x is staged into LDS with the Tensor Data Mover (tensor_load_to_lds,
// TENSORcnt) when the builtin is available; plain loads otherwise.
// Toolchain probe: this environment's builtin is the 6-arg (clang-23) form.
// ---------------------------------------------------------------------------

typedef float v2f __attribute__((ext_vector_type(2)));
typedef float v8f __attribute__((ext_vector_type(8)));
typedef unsigned int u32x4 __attribute__((ext_vector_type(4)));
typedef int i32x4 __attribute__((ext_vector_type(4)));
typedef int i32x8 __attribute__((ext_vector_type(8)));

#if __has_builtin(__builtin_amdgcn_tensor_load_to_lds) && \
    __has_builtin(__builtin_amdgcn_s_wait_tensorcnt)
#define TDM_OK 1
#else
#define TDM_OK 0
#endif

#define T_DIM        256
#define V_DIM        21
#define CO_DIM       64
#define SLAB         5376      // T*V floats of x per (n,ci)
#define ROWS_PER_N   5376      // T*V rows of y3 per n
#define OUT_PER_N    344064    // ROWS_PER_N * CO_DIM
#define NCHAN        1344      // BN channels (ROWS_PER_N/4)
#define NSAMP        16384.f   // N*T samples per BN channel
#define NCHUNK       4
#define J_CHUNK      64
#define ROWS_CHUNK   1344      // J_CHUNK * V_DIM
#define ROWTILES     84        // ROWS_CHUNK / 16

// LDS layout (floats); total 14720 floats = 58,880 B
#define OFF_X64   0        // 252   column sums of x
#define OFF_CW    256      // 576   conv_w (192x3)
#define OFF_CB    832      // 192   conv_b
#define OFF_A     1024     // 1323  A (3x21x21)
#define OFF_W     2368     // 192   W_emb (64x3)
#define OFF_BE    2560     // 64    b_emb
#define OFF_PE    2624     // 1344  pos_emb (21x64)
#define OFF_STAT  3968     // 2688  stats accumulators OR scale/shift
#define OFF_S     6656     // 4032  s (chunk)  -- also x-slab staging (5376)
#define OFF_Y2    10688    // 4032  y2 (chunk) -- tail of staging region
#define LDS_FLOATS 14720

__global__ void infogcn_zero_ws(float* __restrict__ ws) {
    for (int i = threadIdx.x; i < 2 * NCHAN; i += 256) ws[i] = 0.f;
}

__global__ void infogcn_finalize(float* __restrict__ ws,
                                 const float* __restrict__ gamma,
                                 const float* __restrict__ beta) {
    int ch = blockIdx.x * blockDim.x + threadIdx.x;
    if (ch < NCHAN) {
        float mean  = ws[ch]         * (1.f / NSAMP);
        float var   = ws[NCHAN + ch] * (1.f / NSAMP) - mean * mean;
        float inv   = rsqrtf(var + 1e-5f);
        float scale = gamma[ch] * inv;
        ws[2 * NCHAN + ch] = scale;
        ws[3 * NCHAN + ch] = beta[ch] - mean * scale;
    }
}

template <bool STATS>
__global__ __launch_bounds__(256)
void infogcn_pass(const float* __restrict__ x,   const float* __restrict__ A,
                  const float* __restrict__ cw,  const float* __restrict__ cb,
                  const float* __restrict__ W,   const float* __restrict__ be,
                  const float* __restrict__ pe,  float* __restrict__ out,
                  float* __restrict__ ws) {
    __shared__ float lds[LDS_FLOATS];
    const int tid  = threadIdx.x;
    const int n    = blockIdx.x;
    const int lane = tid & 31;
    const int wave = tid >> 5;
    const int half = lane >> 4;   // wave32 half: selects K{0,1} vs K{2,3}
    const int l15  = lane & 15;

    // ---- stage parameters into LDS --------------------------------------
    for (int i = tid; i < 576;  i += 256) lds[OFF_CW + i] = cw[i];
    for (int i = tid; i < 192;  i += 256) lds[OFF_CB + i] = cb[i];
    for (int i = tid; i < 1323; i += 256) lds[OFF_A  + i] = A[i];
    for (int i = tid; i < 192;  i += 256) lds[OFF_W  + i] = W[i];
    for (int i = tid; i < 64;   i += 256) lds[OFF_BE + i] = be[i];
    for (int i = tid; i < 1344; i += 256) lds[OFF_PE + i] = pe[i];
    if (STATS) {
        for (int i = tid; i < 2 * NCHAN; i += 256) lds[OFF_STAT + i] = 0.f;
    } else {
        for (int i = tid; i < 2 * NCHAN; i += 256)
            lds[OFF_STAT + i] = ws[2 * NCHAN + i];  // scale(1344) | shift(1344)
    }

    // ---- X64 via TDM-staged x slabs -------------------------------------
    // Stage x[n,ci,:,:] (5376 contiguous floats) into lds[OFF_S..] with one
    // 1D tensor_load_to_lds per slab (wave 0 only; TDM ignores EXEC), then
    // reduce 64-wide t-blocks into X64.
#if TDM_OK
    const unsigned int ldsbase = (unsigned int)(uintptr_t)&lds[0];
#endif
    for (int ci = 0; ci < 3; ++ci) {
        const float* xslab = x + (size_t)(n * 3 + ci) * SLAB;
#if TDM_OK
        if (wave == 0) {
            unsigned long long ga = (unsigned long long)(uintptr_t)xslab;
            u32x4 g0 = {0u, 0u, 0u, 0u};
            g0.x = 1u;                                   // count=1 (user D#)
            g0.y = ldsbase + OFF_S * 4;                  // lds_addr (bytes)
            g0.z = (unsigned int)ga;                     // global_addr[31:0]
            g0.w = (unsigned int)((ga >> 32) & 0x1FFFFFFull)
                 | (2u << 30);                           // addr[56:32], type=2
            i32x8 g1 = {0, 0, 0, 0, 0, 0, 0, 0};
            g1[0] = (int)(2u << 16);                     // data_size = 4B
            g1[1] = (int)((SLAB & 0xFFFFu) << 16);       // tensor_dim0[15:0]
            g1[2] = (int)(1u << 16);                     // dim0[31:16]=0, tensor_dim1=1
            g1[3] = (int)((unsigned)SLAB << 16);         // tile_dim0 = 5376
            g1[4] = 1;                                   // tile_dim1=1, tile_dim2=0
            g1[5] = SLAB;                                // tensor_dim0_stride
            i32x4 gz4 = {0, 0, 0, 0};                    // groups 2/3 unused (<=2D)
            i32x8 gz8 = {0, 0, 0, 0, 0, 0, 0, 0};
            __builtin_amdgcn_tensor_load_to_lds(g0, g1, gz4, gz4, gz8, 0);
            __builtin_amdgcn_s_wait_tensorcnt((short)0); // TENSORcnt == 0
        }
#else
        for (int i = tid; i < SLAB; i += 256) lds[OFF_S + i] = xslab[i];
#endif
        __syncthreads();
        if (tid < 84) {
            int b = tid / 21, v = tid % 21;
            const float* p = &lds[OFF_S + b * 64 * 21 + v];
            float acc = 0.f;
#pragma unroll 8
            for (int i = 0; i < 64; ++i) acc += p[i * 21];
            lds[OFF_X64 + (ci * 4 + b) * 21 + v] = acc;
        }
        __syncthreads();
    }

    for (int jc = 0; jc < NCHUNK; ++jc) {
        __syncthreads();
        // ---- s[jl,k,v] for this chunk (4032 values) ---------------------
        for (int idx = tid; idx < 4032; idx += 256) {
            int jl = idx / 63, r = idx % 63;
            int k = r / 21, v = r % 21;
            int u  = 3 * (jc * J_CHUNK + jl) + k;
            int kc = u >> 2, tb = u & 3;
            float sv = lds[OFF_CW + kc * 3 + 0] * lds[OFF_X64 + (0 * 4 + tb) * 21 + v]
                     + lds[OFF_CW + kc * 3 + 1] * lds[OFF_X64 + (1 * 4 + tb) * 21 + v]
                     + lds[OFF_CW + kc * 3 + 2] * lds[OFF_X64 + (2 * 4 + tb) * 21 + v]
                     + 64.f * lds[OFF_CB + kc];
            lds[OFF_S + (jl * 3 + k) * 21 + v] = sv;
        }
        __syncthreads();
        // ---- y2[row,k] = s[jl,k,:] . A[k,:,w]  (4032 values) -----------
        for (int idx = tid; idx < 4032; idx += 256) {
            int jl = idx / 63, r = idx % 63;
            int w = r / 3, k = r % 3;
            const float* sp = &lds[OFF_S + (jl * 3 + k) * 21];
            const float* ap = &lds[OFF_A + k * 441 + w];
            float acc = 0.f;
#pragma unroll
            for (int v = 0; v < 21; ++v) acc += sp[v] * ap[v * 21];
            lds[OFF_Y2 + (jl * 21 + w) * 3 + k] = acc;
        }
        __syncthreads();
        // ---- WMMA: y3 = y2(16x4, K=3 padded) @ W_emb^T(4x16) + bias ----
        // uniform per-wave loop -> EXEC all-1s for every v_wmma
        for (int t = wave; t < ROWTILES * 4; t += 8) {
            int rowTile = t >> 2, coTile = t & 3;
            int rowA = rowTile * 16 + l15;     // A: both halves M=0..15
            int co   = coTile  * 16 + l15;
            v2f a, b;
            a.x = lds[OFF_Y2 + rowA * 3 + 2 * half];          // K = 0 or 2
            a.y = half ? 0.f : lds[OFF_Y2 + rowA * 3 + 1];    // K = 1 or 3(pad)
            b.x = lds[OFF_W + co * 3 + 2 * half];             // B[k,co]=W_emb[co,k]
            b.y = half ? 0.f : lds[OFF_W + co * 3 + 1];
            v8f c;
#pragma unroll
            for (int r = 0; r < 8; ++r) {
                int rowLocal = rowTile * 16 + r + 8 * half;   // C/D: M=r / r+8
                int w = (jc * ROWS_CHUNK + rowLocal) % 21;
                c[r] = lds[OFF_BE + co] + lds[OFF_PE + w * 64 + co];
            }
            v8f d = __builtin_amdgcn_wmma_f32_16x16x4_f32(
                false, a, false, b, (short)0, c, false, false);

            if (STATS) {
                // BN channel = rowGlobal>>2: aligned with D-fragment halves.
                float p0 = d[0] + d[1] + d[2] + d[3];
                float p1 = d[4] + d[5] + d[6] + d[7];
                float q0 = d[0]*d[0] + d[1]*d[1] + d[2]*d[2] + d[3]*d[3];
                float q1 = d[4]*d[4] + d[5]*d[5] + d[6]*d[6] + d[7]*d[7];
#pragma unroll
                for (int off = 1; off < 16; off <<= 1) {   // stays within half
                    p0 += __shfl_xor(p0, off, 32);
                    p1 += __shfl_xor(p1, off, 32);
                    q0 += __shfl_xor(q0, off, 32);
                    q1 += __shfl_xor(q1, off, 32);
                }
                if (l15 == 0) {
                    int chBase = ((jc * ROWS_CHUNK + rowTile * 16) >> 2) + half * 2;
                    atomicAdd(&lds[OFF_STAT + chBase],             p0);
                    atomicAdd(&lds[OFF_STAT + chBase + 1],         p1);
                    atomicAdd(&lds[OFF_STAT + NCHAN + chBase],     q0);
                    atomicAdd(&lds[OFF_STAT + NCHAN + chBase + 1], q1);
                }
            } else {
#pragma unroll
                for (int r = 0; r < 8; ++r) {
                    int rowGlobal = jc * ROWS_CHUNK + rowTile * 16 + r + 8 * half;
                    int ch = rowGlobal >> 2;
                    float val = d[r] * lds[OFF_STAT + ch]
                              + lds[OFF_STAT + NCHAN + ch];
                    // streamed 88 MB, never re-read -> non-temporal store
                    __builtin_nontemporal_store(
                        val, &out[(size_t)n * OUT_PER_N + rowGlobal * 64 + co]);
                }
            }
        }
    }

    if (STATS) {
        __syncthreads();
        for (int i = tid; i < 2 * NCHAN; i += 256)
            atomicAdd(&ws[i], lds[OFF_STAT + i]);
    }
}

extern "C" void kernel_launch(void* const* d_in, const int* in_sizes, int n_in,
                              void* d_out, int out_size, void* d_ws, size_t ws_size,
                              hipStream_t stream) {
    (void)in_sizes; (void)n_in; (void)out_size; (void)ws_size;
    const float* x     = (const float*)d_in[0];
    const float* A     = (const float*)d_in[1];
    const float* cw    = (const float*)d_in[2];
    const float* cb    = (const float*)d_in[3];
    const float* W     = (const float*)d_in[4];
    const float* be    = (const float*)d_in[5];
    const float* pe    = (const float*)d_in[6];
    const float* gamma = (const float*)d_in[7];
    const float* beta  = (const float*)d_in[8];
    float* out = (float*)d_out;
    float* ws  = (float*)d_ws;   // [sum 1344 | sumsq 1344 | scale 1344 | shift 1344]

    infogcn_zero_ws<<<1, 256, 0, stream>>>(ws);
    infogcn_pass<true><<<64, 256, 0, stream>>>(x, A, cw, cb, W, be, pe, nullptr, ws);
    infogcn_finalize<<<6, 256, 0, stream>>>(ws, gamma, beta);
    infogcn_pass<false><<<64, 256, 0, stream>>>(x, A, cw, cb, W, be, pe, out, ws);
}